// AdaptiveIFSNet_80556406604127
// MI455X (gfx1250) — compile-verified
//
#include <hip/hip_runtime.h>
#include <math.h>

typedef __attribute__((ext_vector_type(2))) float v2f;
typedef __attribute__((ext_vector_type(8))) float v8f;

#define KIFS 10
#define NHID 32
#define TWO_PI_F 6.28318530717958647692f

// ---------------------------------------------------------------------------
// Setup kernel: one wave32.
//  - gate MLP layer1 via 4x V_WMMA_F32_16X16X4_F32 (16x8 @ 8x32, padded)
//  - layer2 (32->1) via LDS transpose + per-lane dot
//  - analytic SVD clamp: w = R1 * diag(min(sig,0.999)) * R2 * diag(sign d)
//  - serial K=10 count / cumsum logic on lane 0
// Writes to workspace:
//  tw[k*8 + 0..3] = w00,w01,w10,w11 ; tw[k*8+4..5] = b ; ti[k]=csum ; ti[16+k]=start
// ---------------------------------------------------------------------------
__global__ __launch_bounds__(32) void ifs_setup(
    const float* __restrict__ ifs_w, const float* __restrict__ ifs_b,
    const float* __restrict__ gw1,   const float* __restrict__ gb1,
    const float* __restrict__ gw2,   const float* __restrict__ gb2,
    float* __restrict__ tw, int* __restrict__ ti, int N)
{
    __shared__ float Hs[16 * NHID];
    __shared__ float gate_s[16];
    __shared__ float w_s[16 * 4];
    __shared__ float det_s[16];

    const int lane = threadIdx.x;      // 0..31
    const int m    = lane & 15;        // row (A/M) or column (B/N) index
    const int hi   = lane >> 4;        // upper lane half

    // ---- A operand: ifs_w padded [16 x 8], two K-chunks of 4 ----
    // f32 16x4 A layout: lanes 0-15 hold (K=0,K=1) in (v0,v1); lanes 16-31 hold (K=2,K=3)
    v2f A0, A1;
    A0.x = (m < KIFS) ? ifs_w[m * 6 + (hi ? 2 : 0)] : 0.f;
    A0.y = (m < KIFS) ? ifs_w[m * 6 + (hi ? 3 : 1)] : 0.f;
    A1.x = (m < KIFS && !hi) ? ifs_w[m * 6 + 4] : 0.f;   // K=4 (lo) / K=6 pad (hi)
    A1.y = (m < KIFS && !hi) ? ifs_w[m * 6 + 5] : 0.f;   // K=5 (lo) / K=7 pad (hi)

    // ---- B operand: gw1 padded [8 x 32], two N-tiles of 16 ----
    // f32 4x16 B layout (mirror of A): lane n = lane&15; lanes 0-15 hold K=0,1; 16-31 hold K=2,3
    const int nL = m, nH = m + 16;
    v2f B0L, B1L, B0H, B1H;
    B0L.x = gw1[(hi ? 2 : 0) * NHID + nL];
    B0L.y = gw1[(hi ? 3 : 1) * NHID + nL];
    B1L.x = hi ? 0.f : gw1[4 * NHID + nL];
    B1L.y = hi ? 0.f : gw1[5 * NHID + nL];
    B0H.x = gw1[(hi ? 2 : 0) * NHID + nH];
    B0H.y = gw1[(hi ? 3 : 1) * NHID + nH];
    B1H.x = hi ? 0.f : gw1[4 * NHID + nH];
    B1H.y = hi ? 0.f : gw1[5 * NHID + nH];

    v8f accL = {0.f,0.f,0.f,0.f,0.f,0.f,0.f,0.f};
    v8f accH = {0.f,0.f,0.f,0.f,0.f,0.f,0.f,0.f};
    accL = __builtin_amdgcn_wmma_f32_16x16x4_f32(false, A0, false, B0L, (short)0, accL, false, false);
    accL = __builtin_amdgcn_wmma_f32_16x16x4_f32(false, A1, false, B1L, (short)0, accL, false, false);
    accH = __builtin_amdgcn_wmma_f32_16x16x4_f32(false, A0, false, B0H, (short)0, accH, false, false);
    accH = __builtin_amdgcn_wmma_f32_16x16x4_f32(false, A1, false, B1H, (short)0, accH, false, false);

    // bias + relu, scatter to LDS using C/D layout (VGPR v -> rows v / v+8)
    const float bL = gb1[nL], bH = gb1[nH];
#pragma unroll
    for (int v = 0; v < 8; ++v) {
        const int r = v + (hi ? 8 : 0);
        float hl = accL[v] + bL; hl = hl > 0.f ? hl : 0.f;
        float hh = accH[v] + bH; hh = hh > 0.f ? hh : 0.f;
        Hs[r * NHID + nL] = hl;
        Hs[r * NHID + nH] = hh;
    }
    __syncthreads();

    if (lane < KIFS) {
        // layer 2: 32 -> 1, sigmoid gate
        float g = gb2[0];
#pragma unroll
        for (int j = 0; j < NHID; ++j) g += Hs[lane * NHID + j] * gw2[j];
        gate_s[lane] = 1.0f / (1.0f + expf(-g));

        // build contractive 2x2: R1 * diag(sa,sb) * R2 * diag(e1,e2)
        const float* P = ifs_w + lane * 6;
        const float t1 = P[0] * TWO_PI_F, t2 = P[1] * TWO_PI_F;
        const float c1 = cosf(t1), s1 = sinf(t1);
        const float c2 = cosf(t2), s2 = sinf(t2);
        const float sa = fminf(1.0f / (1.0f + expf(-P[2])), 0.999f);
        const float sb = fminf(1.0f / (1.0f + expf(-P[3])), 0.999f);
        const float e1 = (P[4] > 0.f) ? 1.f : ((P[4] < 0.f) ? -1.f : 0.f);
        const float e2 = (P[5] > 0.f) ? 1.f : ((P[5] < 0.f) ? -1.f : 0.f);
        const float w00 = ( sa * c1 * c2 - sb * s1 * s2) * e1;
        const float w01 = (-sa * c1 * s2 - sb * s1 * c2) * e2;
        const float w10 = ( sa * s1 * c2 + sb * c1 * s2) * e1;
        const float w11 = (-sa * s1 * s2 + sb * c1 * c2) * e2;
        w_s[lane * 4 + 0] = w00; w_s[lane * 4 + 1] = w01;
        w_s[lane * 4 + 2] = w10; w_s[lane * 4 + 3] = w11;
        det_s[lane] = fabsf(w00 * w11 - w01 * w10);
    }
    __syncthreads();

    if (lane == 0) {
        // mask + fallback(argmax, first occurrence)
        bool any = false; float gmax = -1e30f; int amax = 0;
        for (int k = 0; k < KIFS; ++k) {
            const float g = gate_s[k];
            if (g > 0.1f) any = true;
            if (g > gmax) { gmax = g; amax = k; }
        }
        float dets[KIFS]; int mask[KIFS]; float dsum = 0.f;
        for (int k = 0; k < KIFS; ++k) {
            mask[k] = any ? (gate_s[k] > 0.1f ? 1 : 0) : (k == amax ? 1 : 0);
            dets[k] = mask[k] ? det_s[k] : 0.f;
            dsum += dets[k];
        }
        int counts[KIFS]; long total = 0; int last_valid = 0;
        for (int k = 0; k < KIFS; ++k) {
            const float pr = dets[k] / (dsum + 1e-8f);
            int c = 0;
            if (mask[k]) {
                c = (int)rintf(pr * (float)N);   // round-half-even, matches jnp.round
                if (c < 1) c = 1;
                last_valid = k;
            }
            counts[k] = c; total += c;
        }
        counts[last_valid] += (int)((long)N - total);
        int run = 0;
        for (int k = 0; k < KIFS; ++k) {
            const int start = run;
            run += counts[k];
            ti[k]      = run;    // inclusive cumsum
            ti[16 + k] = start;  // segment start = csum - counts
            tw[k * 8 + 0] = w_s[k * 4 + 0];
            tw[k * 8 + 1] = w_s[k * 4 + 1];
            tw[k * 8 + 2] = w_s[k * 4 + 2];
            tw[k * 8 + 3] = w_s[k * 4 + 3];
            tw[k * 8 + 4] = ifs_b[k * 2 + 0];
            tw[k * 8 + 5] = ifs_b[k * 2 + 1];
            tw[k * 8 + 6] = 0.f;
            tw[k * 8 + 7] = 0.f;
        }
    }
}

// ---------------------------------------------------------------------------
// Streaming kernel: HBM-bound. 2 points per thread.
// loads: 2x global_load_b64 (float2), store: 1x global_store_b128 (float4)
// cloud search: 10 register compares against csum table (LDS-broadcast once)
// ---------------------------------------------------------------------------
__global__ __launch_bounds__(256) void ifs_apply(
    const float* __restrict__ p_in,
    const float* __restrict__ tw,
    const int*   __restrict__ ti,
    float* __restrict__ out, int N)
{
    __shared__ float sw[80];
    __shared__ int   si[26];
    const int t = threadIdx.x;
    if (t < 80) sw[t] = tw[t];
    if (t < 26) si[t] = ti[t];
    __syncthreads();

    int cs[KIFS];
#pragma unroll
    for (int k = 0; k < KIFS; ++k) cs[k] = si[k];

    const int gid = blockIdx.x * 256 + t;
    const int j0  = gid * 2;
    if (j0 >= N) return;

    float4 o;
#pragma unroll
    for (int u = 0; u < 2; ++u) {
        const int j = j0 + u;
        int c = 0;
#pragma unroll
        for (int k = 0; k < KIFS; ++k) c += (cs[k] <= j) ? 1 : 0;  // searchsorted right
        c = c > (KIFS - 1) ? (KIFS - 1) : c;
        int within = j - si[16 + c];
        within = within < 0 ? 0 : (within > N - 1 ? N - 1 : within);
        const float2 p = ((const float2*)p_in)[within];
        const float* W = &sw[c * 8];
        const float ox = fmaf(W[0], p.x, fmaf(W[1], p.y, W[4]));
        const float oy = fmaf(W[2], p.x, fmaf(W[3], p.y, W[5]));
        if (u == 0) { o.x = ox; o.y = oy; } else { o.z = ox; o.w = oy; }
    }
    ((float4*)out)[gid] = o;
}

extern "C" void kernel_launch(void* const* d_in, const int* in_sizes, int n_in,
                              void* d_out, int out_size, void* d_ws, size_t ws_size,
                              hipStream_t stream)
{
    const float* p_in  = (const float*)d_in[0];
    const float* ifs_w = (const float*)d_in[1];
    const float* ifs_b = (const float*)d_in[2];
    const float* gw1   = (const float*)d_in[3];
    const float* gb1   = (const float*)d_in[4];
    const float* gw2   = (const float*)d_in[5];
    const float* gb2   = (const float*)d_in[6];
    float* out = (float*)d_out;
    const int N = in_sizes[0] / 2;   // number of 2-D points (8388608)

    float* tw = (float*)d_ws;                    // 80 floats: per-cloud W/b table
    int*   ti = (int*)((char*)d_ws + 512);       // csum[0..9], start[16..25]

    ifs_setup<<<1, 32, 0, stream>>>(ifs_w, ifs_b, gw1, gb1, gw2, gb2, tw, ti, N);

    const int nthreads = (N + 1) / 2;            // 2 points per thread
    const int blocks   = (nthreads + 255) / 256;
    ifs_apply<<<blocks, 256, 0, stream>>>(p_in, tw, ti, out, N);
}